// VectorQuantizer_21620865368701
// MI455X (gfx1250) — compile-verified
//
#include <hip/hip_runtime.h>

#define N_ROWS   16384   // 16*32*32 flattened pixels
#define N_CODES  8192
#define CDIM     64
#define HW       1024    // 32*32
#define MT       2       // 16-row tiles per block
#define ROWS_PER_BLOCK (16*MT)

typedef float v2f __attribute__((ext_vector_type(2)));
typedef float v8f __attribute__((ext_vector_type(8)));

// ---------------- kernel 1: half squared norms of codebook rows ----------------
__global__ void vq_cnorm(const float* __restrict__ cb, float* __restrict__ hnorm) {
    int k = blockIdx.x * blockDim.x + threadIdx.x;
    if (k >= N_CODES) return;
    const float4* r = (const float4*)(cb + (size_t)k * CDIM);
    float s = 0.f;
#pragma unroll
    for (int i = 0; i < CDIM / 4; ++i) {
        float4 v = r[i];
        s += v.x * v.x + v.y * v.y + v.z * v.z + v.w * v.w;
    }
    hnorm[k] = 0.5f * s;
}

// ---------------- kernel 2: fused WMMA GEMM + argmin ----------------
// flat[n][c] = hidden[b*64*1024 + c*1024 + hw], n = b*1024 + hw
// score s = x.c ; argmin over codes of (0.5*||c||^2 - s)  (== argmin of d2)
__global__ __launch_bounds__(256) void vq_argmin(
    const float* __restrict__ hidden, const float* __restrict__ cb,
    const float* __restrict__ hnorm, int* __restrict__ idx_out,
    float* __restrict__ idx_out_f)
{
    __shared__ float s_val[ROWS_PER_BLOCK][8];
    __shared__ int   s_idx[ROWS_PER_BLOCK][8];

    const int lane = threadIdx.x & 31;
    const int wave = threadIdx.x >> 5;
    const int lm   = lane & 15;   // row (A) / column-code (B,D) within tile
    const int half = lane >> 4;   // selects K-pair {0,1} vs {2,3}

    const int m0 = blockIdx.x * ROWS_PER_BLOCK;

    // A-tiles in registers: fp32 16x4 layout -> lane holds A[m][2*half + j],
    // stepping base K by 4 per WMMA. a[t][kk] covers c = 4*kk + 2*half + {0,1}.
    v2f a[MT][16];
#pragma unroll
    for (int t = 0; t < MT; ++t) {
        int n = m0 + t * 16 + lm;
        const float* hb = hidden + (size_t)(n >> 10) * (CDIM * HW) + (n & (HW - 1));
#pragma unroll
        for (int kk = 0; kk < 16; ++kk) {
            int c = 4 * kk + 2 * half;
            a[t][kk].x = hb[(size_t)c * HW];
            a[t][kk].y = hb[(size_t)(c + 1) * HW];
        }
    }

    float bestv[MT][8];
    int   besti[MT][8];
#pragma unroll
    for (int t = 0; t < MT; ++t)
#pragma unroll
        for (int i = 0; i < 8; ++i) { bestv[t][i] = 3.4e38f; besti[t][i] = 0; }

    // each wave scans code tiles wave, wave+8, ... (uniform per wave; EXEC all 1s)
    for (int ct = wave; ct < N_CODES / 16; ct += 8) {
        const int code = ct * 16 + lm;
        const float* crow = cb + (size_t)code * CDIM + 2 * half;
        const float hc = hnorm[code];

        v2f b[16];
#pragma unroll
        for (int kk = 0; kk < 16; ++kk)
            b[kk] = *(const v2f*)(crow + 4 * kk);   // 8B-aligned b64 loads

        v8f acc[MT] = {};
#pragma unroll
        for (int kk = 0; kk < 16; ++kk) {
#pragma unroll
            for (int t = 0; t < MT; ++t) {
                acc[t] = __builtin_amdgcn_wmma_f32_16x16x4_f32(
                    false, a[t][kk], false, b[kk], (short)0, acc[t], false, false);
            }
        }

        // D layout: VGPR i -> row (i + 8*half), column code = ct*16 + lm
#pragma unroll
        for (int t = 0; t < MT; ++t)
#pragma unroll
            for (int i = 0; i < 8; ++i) {
                float v = hc - acc[t][i];
                if (v < bestv[t][i]) { bestv[t][i] = v; besti[t][i] = code; }
            }
    }

    // reduce across the 16 lanes of each half (xor masks 8..1 stay within halves)
#pragma unroll
    for (int t = 0; t < MT; ++t)
#pragma unroll
        for (int i = 0; i < 8; ++i) {
            float v = bestv[t][i]; int ix = besti[t][i];
#pragma unroll
            for (int m = 8; m >= 1; m >>= 1) {
                float ov = __shfl_xor(v, m, 32);
                int   oi = __shfl_xor(ix, m, 32);
                if (ov < v || (ov == v && oi < ix)) { v = ov; ix = oi; }
            }
            if (lm == 0) {
                int row = t * 16 + i + 8 * half;
                s_val[row][wave] = v;
                s_idx[row][wave] = ix;
            }
        }
    __syncthreads();

    if (threadIdx.x < ROWS_PER_BLOCK) {
        int row = threadIdx.x;
        float v = s_val[row][0]; int ix = s_idx[row][0];
#pragma unroll
        for (int w = 1; w < 8; ++w) {
            float ov = s_val[row][w]; int oi = s_idx[row][w];
            if (ov < v || (ov == v && oi < ix)) { v = ov; ix = oi; }
        }
        int n = m0 + row;
        idx_out[n]   = ix;
        idx_out_f[n] = (float)ix;
    }
}

// ---------------- kernel 3: gather z_q (B,C,H,W layout) + partial loss ----------------
__global__ void vq_gather_loss(const float* __restrict__ hidden,
                               const float* __restrict__ cb,
                               const int* __restrict__ idx,
                               float* __restrict__ zq_out,
                               float* __restrict__ partial)
{
    __shared__ float red[256];
    const int tid = threadIdx.x;
    float local = 0.f;
    size_t base = (size_t)blockIdx.x * 1024;
#pragma unroll
    for (int r = 0; r < 4; ++r) {
        size_t e = base + (size_t)r * 256 + tid;    // e = b*65536 + c*1024 + hw
        int b  = (int)(e >> 16);
        int c  = (int)((e >> 10) & 63);
        int hw = (int)(e & 1023);
        int n  = b * HW + hw;
        float z = cb[(size_t)idx[n] * CDIM + c];
        float h = hidden[e];                        // same (B,C,H,W) layout
        float d = z - h;
        zq_out[e] = z;                              // straight-through output == z_q
        local += d * d;
    }
    red[tid] = local;
    __syncthreads();
    for (int s = 128; s > 0; s >>= 1) {
        if (tid < s) red[tid] += red[tid + s];
        __syncthreads();
    }
    if (tid == 0) partial[blockIdx.x] = red[0];
}

// ---------------- kernel 4: deterministic final loss reduction ----------------
__global__ void vq_loss_final(const float* __restrict__ partial, float* __restrict__ loss_out) {
    __shared__ float red[256];
    float s = 0.f;
    for (int i = threadIdx.x; i < 1024; i += 256) s += partial[i];
    red[threadIdx.x] = s;
    __syncthreads();
    for (int st = 128; st > 0; st >>= 1) {
        if (threadIdx.x < st) red[threadIdx.x] += red[threadIdx.x + st];
        __syncthreads();
    }
    if (threadIdx.x == 0)
        loss_out[0] = 1.25f * red[0] / (float)((size_t)N_ROWS * CDIM);
}

extern "C" void kernel_launch(void* const* d_in, const int* in_sizes, int n_in,
                              void* d_out, int out_size, void* d_ws, size_t ws_size,
                              hipStream_t stream) {
    const float* hidden = (const float*)d_in[0];   // (16,64,32,32) fp32
    const float* cb     = (const float*)d_in[1];   // (8192,64) fp32

    float* out    = (float*)d_out;
    float* zq_out = out;                           // 16*64*32*32 = 1048576
    float* idx_f  = out + 1048576;                 // 16*1024 = 16384
    float* loss   = out + 1048576 + 16384;         // 1

    char*  ws    = (char*)d_ws;
    float* hnorm = (float*)ws;                         // 8192 floats
    int*   idx   = (int*)(ws + 8192 * 4);              // 16384 ints
    float* part  = (float*)(ws + 8192 * 4 + 16384 * 4);// 1024 floats

    vq_cnorm     <<<N_CODES / 256, 256, 0, stream>>>(cb, hnorm);
    vq_argmin    <<<N_ROWS / ROWS_PER_BLOCK, 256, 0, stream>>>(hidden, cb, hnorm, idx, idx_f);
    vq_gather_loss<<<1024, 256, 0, stream>>>(hidden, cb, idx, zq_out, part);
    vq_loss_final <<<1, 256, 0, stream>>>(part, loss);
}